// GlobalDistNet_77360950936271
// MI455X (gfx1250) — compile-verified
//
#include <hip/hip_runtime.h>
#include <math.h>

typedef __attribute__((ext_vector_type(2))) float v2f;
typedef __attribute__((ext_vector_type(8))) float v8f;

#define CH 128

// ---------------------------------------------------------------- WMMA helper
__device__ __forceinline__ v8f wmma_f32(v2f a, v2f b, v8f c) {
  // V_WMMA_F32_16X16X4_F32 : D = A(16x4) * B(4x16) + C(16x16), all fp32
  return __builtin_amdgcn_wmma_f32_16x16x4_f32(
      /*neg_a=*/false, a, /*neg_b=*/false, b,
      /*c_mod=*/(short)0, c, /*reuse_a=*/false, /*reuse_b=*/false);
}

__device__ __forceinline__ float lrelu01(float v) { return v > 0.f ? v : 0.01f * v; }
__device__ __forceinline__ float lrelu20(float v) { return v > 0.f ? v : 0.2f * v; }

__device__ __forceinline__ void atomicMaxF(float* addr, float val) {
  unsigned int* ua = (unsigned int*)addr;
  unsigned int cur = *((volatile unsigned int*)ua);
  while (__uint_as_float(cur) < val) {
    unsigned int prev = atomicCAS(ua, cur, __float_as_uint(val));
    if (prev == cur) break;
    cur = prev;
  }
}

// --------------------------------------------- async global->LDS copy (CDNA5)
// Issues GLOBAL_LOAD_ASYNC_TO_LDS_B128 (ASYNCcnt); 16 instructions per wave
// move one 64KB W block with 256 threads (16B per lane per instruction).
__device__ __forceinline__ void async_copy_w64k(const float* __restrict__ src,
                                                float* __restrict__ dstLds) {
#pragma unroll
  for (int j = 0; j < 16; ++j) {
    int idx = (threadIdx.x + j * 256) * 4;  // float index, 16B per lane
    unsigned int dst = (unsigned int)(uintptr_t)(dstLds + idx);
    unsigned long long s = (unsigned long long)(uintptr_t)(src + idx);
    asm volatile("global_load_async_to_lds_b128 %0, %1, off"
                 :: "v"(dst), "v"(s) : "memory");
  }
}
__device__ __forceinline__ void wait_async16() {
  asm volatile("s_wait_asynccnt 0x10" ::: "memory");
}
__device__ __forceinline__ void wait_async0() {
  asm volatile("s_wait_asynccnt 0x0" ::: "memory");
}

// ---------------------------------------------------------------- utilities
__global__ void k_zero(float* __restrict__ p, size_t cnt) {
  size_t i = (size_t)blockIdx.x * blockDim.x + threadIdx.x;
  if (i < cnt) p[i] = 0.f;
}
__global__ void k_fill(float* __restrict__ p, size_t cnt, float v) {
  size_t i = (size_t)blockIdx.x * blockDim.x + threadIdx.x;
  if (i < cnt) p[i] = v;
}

// ---------------------------------------------------------------- degree / norm
__global__ void k_deg(const int* __restrict__ ei, const float* __restrict__ w,
                      float* __restrict__ deg, int E, int n) {
  int e = blockIdx.x * blockDim.x + threadIdx.x;
  if (e >= E + n) return;
  int c; float wt;
  if (e < E) { c = ei[(size_t)E + e]; wt = w[e]; }
  else       { c = e - E;             wt = 1.f;  }
  atomicAdd(&deg[c], wt);
}
__global__ void k_dinv(const float* __restrict__ deg, float* __restrict__ dinv, int n) {
  int i = blockIdx.x * blockDim.x + threadIdx.x;
  if (i < n) dinv[i] = deg[i] > 0.f ? rsqrtf(deg[i]) : 0.f;
}

// ------------------------------------------------- C-tile store (16x128 slab)
__device__ __forceinline__ void store_ctile(float* __restrict__ T, const v8f* acc,
                                            int mbase, int ncol, int n, bool full) {
  if (full) {
#pragma unroll
    for (int nt = 0; nt < 8; ++nt)
#pragma unroll
      for (int r = 0; r < 8; ++r)
        T[(size_t)(mbase + r) * CH + nt * 16 + ncol] = acc[nt][r];
  } else {
#pragma unroll
    for (int nt = 0; nt < 8; ++nt)
#pragma unroll
      for (int r = 0; r < 8; ++r) {
        int orow = mbase + r;
        if (orow < n) T[(size_t)orow * CH + nt * 16 + ncol] = acc[nt][r];
      }
  }
}

// ---------------------------------------------------------------- GEMM [n,128]x[128,128]
__global__ __launch_bounds__(256) void k_gemm128(const float* __restrict__ X,
                                                 const float* __restrict__ W,
                                                 float* __restrict__ T, int n) {
  __shared__ float lw[CH * CH];  // 64 KB
  for (int i = threadIdx.x; i < CH * CH / 4; i += 256)
    reinterpret_cast<float4*>(lw)[i] = reinterpret_cast<const float4*>(W)[i];
  __syncthreads();

  const int wave = threadIdx.x >> 5;
  const int lane = threadIdx.x & 31;
  const int m    = lane & 15;
  const int kh   = (lane >> 4) << 1;  // 0 or 2
  int row  = blockIdx.x * 128 + wave * 16 + m;
  int rowc = row < n ? row : n - 1;
  const float* xr = X + (size_t)rowc * CH;

  v8f z = {0.f, 0.f, 0.f, 0.f, 0.f, 0.f, 0.f, 0.f};
  v8f acc[8] = {z, z, z, z, z, z, z, z};

  for (int k = 0; k < 32; ++k) {
    int kk = k * 4 + kh;
    v2f a; a.x = xr[kk]; a.y = xr[kk + 1];
#pragma unroll
    for (int nt = 0; nt < 8; ++nt) {
      int nc = nt * 16 + m;
      v2f b; b.x = lw[kk * CH + nc]; b.y = lw[(kk + 1) * CH + nc];
      acc[nt] = wmma_f32(a, b, acc[nt]);
    }
  }
  const int mbase = blockIdx.x * 128 + wave * 16 + ((lane >> 4) << 3);
  const bool full = blockIdx.x * 128 + 128 <= n;
  store_ctile(T, acc, mbase, lane & 15, n, full);
}

// ------------------------------------------------- input GEMM with fused emb gather
// x0 = [emb[poi_ids] (32*128 cols) | dist (32 cols)],  T = x0 @ W  (W: [4128,128])
// W blocks double-buffered in LDS via async global->LDS loads (ASYNCcnt).
__global__ __launch_bounds__(256) void k_gemm_input(const float* __restrict__ feat,
                                                    const float* __restrict__ emb,
                                                    const float* __restrict__ W,
                                                    float* __restrict__ T,
                                                    int n, int emb_rows) {
  __shared__ float lw[2][CH * CH];  // 2 x 64 KB double buffer
  const int wave = threadIdx.x >> 5;
  const int lane = threadIdx.x & 31;
  const int m    = lane & 15;
  const int kh   = (lane >> 4) << 1;
  int row  = blockIdx.x * 128 + wave * 16 + m;
  int rowc = row < n ? row : n - 1;

  v8f z = {0.f, 0.f, 0.f, 0.f, 0.f, 0.f, 0.f, 0.f};
  v8f acc[8] = {z, z, z, z, z, z, z, z};

  // prefetch W block 0
  async_copy_w64k(W, lw[0]);

  for (int jb = 0; jb < 32; ++jb) {
    if (jb + 1 < 32) {
      async_copy_w64k(W + (size_t)(jb + 1) * CH * CH, lw[(jb + 1) & 1]);
      wait_async16();  // only the 16 newest in flight => block jb complete
    } else {
      wait_async0();
    }
    __syncthreads();  // async data visible to all waves

    const float* lb = lw[jb & 1];
    int pid = (int)feat[(size_t)rowc * 64 + jb];
    pid = pid < 0 ? 0 : (pid >= emb_rows ? emb_rows - 1 : pid);
    const float* ar = emb + (size_t)pid * CH;

    for (int k = 0; k < 32; ++k) {
      int kk = k * 4 + kh;
      v2f a; a.x = ar[kk]; a.y = ar[kk + 1];
#pragma unroll
      for (int nt = 0; nt < 8; ++nt) {
        int nc = nt * 16 + m;
        v2f b; b.x = lb[kk * CH + nc]; b.y = lb[(kk + 1) * CH + nc];
        acc[nt] = wmma_f32(a, b, acc[nt]);
      }
    }
    __syncthreads();  // all waves done with buf[jb&1] before it is overwritten
  }

  // dist block: K rows 4096..4127 (16 KB, plain cooperative load)
  for (int i = threadIdx.x; i < 32 * CH / 4; i += 256)
    reinterpret_cast<float4*>(lw[0])[i] =
        reinterpret_cast<const float4*>(W + (size_t)4096 * CH)[i];
  __syncthreads();
  for (int k = 0; k < 8; ++k) {
    int kk = k * 4 + kh;
    v2f a;
    a.x = feat[(size_t)rowc * 64 + 32 + kk];
    a.y = feat[(size_t)rowc * 64 + 32 + kk + 1];
#pragma unroll
    for (int nt = 0; nt < 8; ++nt) {
      int nc = nt * 16 + m;
      v2f b; b.x = lw[0][kk * CH + nc]; b.y = lw[0][(kk + 1) * CH + nc];
      acc[nt] = wmma_f32(a, b, acc[nt]);
    }
  }
  const int mbase = blockIdx.x * 128 + wave * 16 + ((lane >> 4) << 3);
  const bool full = blockIdx.x * 128 + 128 <= n;
  store_ctile(T, acc, mbase, lane & 15, n, full);
}

// ---------------------------------------------------------------- aggregations
// one wave per edge, 4 channels per lane
__global__ void k_agg_gcn(const int* __restrict__ ei, const float* __restrict__ w,
                          const float* __restrict__ dinv, const float* __restrict__ T,
                          float* __restrict__ H, int E, int n) {
  size_t g = (size_t)blockIdx.x * blockDim.x + threadIdx.x;
  int edge = (int)(g >> 5);
  int lane = (int)(g & 31);
  if (edge >= E + n) return;
  int r, c; float wt;
  if (edge < E) { r = ei[edge]; c = ei[(size_t)E + edge]; wt = w[edge]; }
  else          { r = c = edge - E;                        wt = 1.f;    }
  float nr = dinv[r] * wt * dinv[c];
  const float* tr = T + (size_t)r * CH;
  float* hr = H + (size_t)c * CH;
#pragma unroll
  for (int j = 0; j < 4; ++j) {
    int ch = lane + 32 * j;
    atomicAdd(&hr[ch], tr[ch] * nr);
  }
}

__global__ void k_agg_scalar(const int* __restrict__ ei, const float* __restrict__ w,
                             const float* __restrict__ dinv, const float* __restrict__ T,
                             float* __restrict__ xg, int E, int n) {
  int edge = blockIdx.x * blockDim.x + threadIdx.x;
  if (edge >= E + n) return;
  int r, c; float wt;
  if (edge < E) { r = ei[edge]; c = ei[(size_t)E + edge]; wt = w[edge]; }
  else          { r = c = edge - E;                        wt = 1.f;    }
  atomicAdd(&xg[c], T[r] * dinv[r] * wt * dinv[c]);
}

// ---------------------------------------------------------------- GAT pieces
__global__ void k_gat_scores(const float* __restrict__ T, const float* __restrict__ as,
                             const float* __restrict__ ad, float* __restrict__ s,
                             float* __restrict__ d, int n) {
  int i = blockIdx.x * blockDim.x + threadIdx.x;
  if (i >= n) return;
  const float* t = T + (size_t)i * CH;
  float a = 0.f, b = 0.f;
  for (int c = 0; c < CH; ++c) { float v = t[c]; a += v * as[c]; b += v * ad[c]; }
  s[i] = a; d[i] = b;
}

__global__ void k_edge_max(const int* __restrict__ ei, const float* __restrict__ s,
                           const float* __restrict__ d, float* __restrict__ emax,
                           int E, int n) {
  int edge = blockIdx.x * blockDim.x + threadIdx.x;
  if (edge >= E + n) return;
  int r, c;
  if (edge < E) { r = ei[edge]; c = ei[(size_t)E + edge]; }
  else          { r = c = edge - E; }
  atomicMaxF(&emax[c], lrelu20(s[r] + d[c]));
}

__global__ void k_edge_exp(const int* __restrict__ ei, const float* __restrict__ s,
                           const float* __restrict__ d, const float* __restrict__ emax,
                           float* __restrict__ esum, float* __restrict__ exb,
                           int E, int n) {
  int edge = blockIdx.x * blockDim.x + threadIdx.x;
  if (edge >= E + n) return;
  int r, c;
  if (edge < E) { r = ei[edge]; c = ei[(size_t)E + edge]; }
  else          { r = c = edge - E; }
  float ex = __expf(lrelu20(s[r] + d[c]) - emax[c]);
  exb[edge] = ex;
  atomicAdd(&esum[c], ex);
}

__global__ void k_agg_att(const int* __restrict__ ei, const float* __restrict__ T,
                          const float* __restrict__ exb, const float* __restrict__ esum,
                          float* __restrict__ H, int E, int n) {
  size_t g = (size_t)blockIdx.x * blockDim.x + threadIdx.x;
  int edge = (int)(g >> 5);
  int lane = (int)(g & 31);
  if (edge >= E + n) return;
  int r, c;
  if (edge < E) { r = ei[edge]; c = ei[(size_t)E + edge]; }
  else          { r = c = edge - E; }
  float alpha = exb[edge] / esum[c];
  const float* tr = T + (size_t)r * CH;
  float* hr = H + (size_t)c * CH;
#pragma unroll
  for (int j = 0; j < 4; ++j) {
    int ch = lane + 32 * j;
    atomicAdd(&hr[ch], tr[ch] * alpha);
  }
}

// ---------------------------------------------------------------- group norm
__global__ void k_gn_stats(const float* __restrict__ H, const float* __restrict__ bias,
                           float* __restrict__ stats, int n) {
  __shared__ float ssum[CH], ssq[CH];
  if (threadIdx.x < CH) { ssum[threadIdx.x] = 0.f; ssq[threadIdx.x] = 0.f; }
  __syncthreads();
  size_t total = (size_t)n * CH;
  for (size_t i = (size_t)blockIdx.x * blockDim.x + threadIdx.x; i < total;
       i += (size_t)gridDim.x * blockDim.x) {
    int c = (int)(i & (CH - 1));
    float v = H[i] + bias[c];
    atomicAdd(&ssum[c], v);
    atomicAdd(&ssq[c], v * v);
  }
  __syncthreads();
  if (threadIdx.x < CH) {
    atomicAdd(&stats[threadIdx.x], ssum[threadIdx.x]);
    atomicAdd(&stats[CH + threadIdx.x], ssq[threadIdx.x]);
  }
}

__global__ void k_gn_update(float* __restrict__ X, const float* __restrict__ H,
                            const float* __restrict__ bias, const float* __restrict__ stats,
                            const float* __restrict__ gw, const float* __restrict__ gb,
                            const float* __restrict__ gms, int n) {
  size_t total = (size_t)n * CH;
  size_t i = (size_t)blockIdx.x * blockDim.x + threadIdx.x;
  if (i >= total) return;
  int c = (int)(i & (CH - 1));
  float fn   = (float)n;
  float mean = stats[c] / fn;
  float m2   = stats[CH + c] / fn;
  float ms   = gms[c];
  // var of (v - ms*mean) = E[v^2] - 2*ms*mean*E[v] + ms^2*mean^2
  float var = m2 - 2.f * ms * mean * mean + ms * ms * mean * mean;
  float inv = rsqrtf(var + 1e-5f);
  float v = H[i] + bias[c];
  float y = gw[c] * (v - ms * mean) * inv + gb[c];
  X[i] += lrelu01(y);
}

// ---------------------------------------------------------------- head / FC
__global__ void k_bias_lrelu(const float* __restrict__ H, const float* __restrict__ b,
                             float* __restrict__ X, size_t total) {
  size_t i = (size_t)blockIdx.x * blockDim.x + threadIdx.x;
  if (i >= total) return;
  X[i] = lrelu01(H[i] + b[i & (CH - 1)]);
}

__global__ void k_dot_out(const float* __restrict__ X, const float* __restrict__ w,
                          float* __restrict__ T, int n) {
  int i = blockIdx.x * blockDim.x + threadIdx.x;
  if (i >= n) return;
  const float* x = X + (size_t)i * CH;
  float a = 0.f;
  for (int c = 0; c < CH; ++c) a += x[c] * w[c];
  T[i] = a;
}

__global__ void k_finalize_xg(float* __restrict__ xg, const float* __restrict__ b, int n) {
  int i = blockIdx.x * blockDim.x + threadIdx.x;
  if (i < n) xg[i] = lrelu01(xg[i] + b[0]);
}

// h[c] = sum_i xg[i] * W1[i,c]  (block = 256 threads = 2 row lanes x 128 channels)
__global__ void k_fc1(const float* __restrict__ xg, const float* __restrict__ W1,
                      float* __restrict__ hacc, int n) {
  __shared__ float part[CH];
  if (threadIdx.x < CH) part[threadIdx.x] = 0.f;
  __syncthreads();
  int c  = threadIdx.x & (CH - 1);
  int rl = threadIdx.x >> 7;  // 0..1
  int base = blockIdx.x * 512;
  int lim  = base + 512 < n ? base + 512 : n;
  float acc = 0.f;
  for (int row = base + rl; row < lim; row += 2)
    acc += xg[row] * W1[(size_t)row * CH + c];
  atomicAdd(&part[c], acc);
  __syncthreads();
  if (threadIdx.x < CH) atomicAdd(&hacc[threadIdx.x], part[threadIdx.x]);
}

__global__ void k_fc2(const float* __restrict__ hacc, const float* __restrict__ b1,
                      const float* __restrict__ W2, const float* __restrict__ b2,
                      float* __restrict__ out, int plen) {
  __shared__ float hh[CH];
  if (threadIdx.x < CH) hh[threadIdx.x] = fmaxf(hacc[threadIdx.x] + b1[threadIdx.x], 0.f);
  __syncthreads();
  int p = blockIdx.x * blockDim.x + threadIdx.x;
  if (p >= plen) return;
  float acc = b2[p];
  for (int c = 0; c < CH; ++c) acc += hh[c] * W2[(size_t)c * plen + p];
  out[p] = fmaxf(acc, 0.f);
}

// ================================================================ launcher
extern "C" void kernel_launch(void* const* d_in, const int* in_sizes, int n_in,
                              void* d_out, int out_size, void* d_ws, size_t ws_size,
                              hipStream_t stream) {
  // pytree flatten order: feature, mask, edge_index, weight, then params dict
  // with keys sorted: cov_in_W, cov_in_b, cov_out_W, cov_out_b, emb, fc_W1,
  // fc_b1, fc_W2, fc_b2, layers[0..3]{att_dst, att_src, gat_W, gat_b, gcn_W,
  // gcn_b, gn_b, gn_ms, gn_w}
  const float* feature   = (const float*)d_in[0];
  const int*   ei        = (const int*)d_in[2];
  const float* ew        = (const float*)d_in[3];
  const float* cov_in_W  = (const float*)d_in[4];
  const float* cov_in_b  = (const float*)d_in[5];
  const float* cov_out_W = (const float*)d_in[6];
  const float* cov_out_b = (const float*)d_in[7];
  const float* emb       = (const float*)d_in[8];
  const float* fc_W1     = (const float*)d_in[9];
  const float* fc_b1     = (const float*)d_in[10];
  const float* fc_W2     = (const float*)d_in[11];
  const float* fc_b2     = (const float*)d_in[12];
  struct Layer { const float *att_dst, *att_src, *gat_W, *gat_b, *gcn_W, *gcn_b, *gn_b, *gn_ms, *gn_w; };
  Layer L[4];
  for (int l = 0; l < 4; ++l) {
    const int b = 13 + 9 * l;
    L[l].att_dst = (const float*)d_in[b + 0];
    L[l].att_src = (const float*)d_in[b + 1];
    L[l].gat_W   = (const float*)d_in[b + 2];
    L[l].gat_b   = (const float*)d_in[b + 3];
    L[l].gcn_W   = (const float*)d_in[b + 4];
    L[l].gcn_b   = (const float*)d_in[b + 5];
    L[l].gn_b    = (const float*)d_in[b + 6];
    L[l].gn_ms   = (const float*)d_in[b + 7];
    L[l].gn_w    = (const float*)d_in[b + 8];
  }

  const int n        = in_sizes[0] / 64;
  const int E        = in_sizes[3];
  const int plen     = out_size;
  const int emb_rows = in_sizes[8] / CH;
  const size_t nCH   = (size_t)n * CH;

  float* ws   = (float*)d_ws;
  float* x    = ws; ws += nCH;
  float* t    = ws; ws += nCH;
  float* h    = ws; ws += nCH;
  float* exb  = ws; ws += (size_t)(E + n);
  float* deg  = ws; ws += n;
  float* dinv = ws; ws += n;
  float* sv   = ws; ws += n;
  float* dv   = ws; ws += n;
  float* emax = ws; ws += n;
  float* esum = ws; ws += n;
  float* xg   = ws; ws += n;
  float* st   = ws; ws += 2 * CH;
  float* hfc  = ws; ws += CH;
  (void)ws_size; (void)n_in;

  const int BLK = 256;
  auto blocks = [](size_t cnt) { return (unsigned)((cnt + 255) / 256); };
  const unsigned mb  = (unsigned)((n + 127) / 128);        // GEMM grid
  const size_t  eth  = (size_t)(E + n) * 32;               // wave-per-edge threads
  const unsigned egB = blocks((size_t)(E + n));            // thread-per-edge grid

  // --- degree / symmetric norm (static over layers) ---
  k_zero<<<blocks(n), BLK, 0, stream>>>(deg, n);
  k_deg<<<egB, BLK, 0, stream>>>(ei, ew, deg, E, n);
  k_dinv<<<blocks(n), BLK, 0, stream>>>(deg, dinv, n);

  // --- input layer: x = lrelu(gcn([emb gather | dist] @ cov_in_W) + b) ---
  k_gemm_input<<<mb, BLK, 0, stream>>>(feature, emb, cov_in_W, t, n, emb_rows);
  k_zero<<<blocks(nCH), BLK, 0, stream>>>(h, nCH);
  k_agg_gcn<<<blocks(eth), BLK, 0, stream>>>(ei, ew, dinv, t, h, E, n);
  k_bias_lrelu<<<blocks(nCH), BLK, 0, stream>>>(h, cov_in_b, x, nCH);

  // --- 4 residual GCN+GAT layers ---
  for (int l = 0; l < 4; ++l) {
    // GCN branch
    k_gemm128<<<mb, BLK, 0, stream>>>(x, L[l].gcn_W, t, n);
    k_zero<<<blocks(nCH), BLK, 0, stream>>>(h, nCH);
    k_agg_gcn<<<blocks(eth), BLK, 0, stream>>>(ei, ew, dinv, t, h, E, n);
    k_zero<<<1, BLK, 0, stream>>>(st, 2 * CH);
    k_gn_stats<<<480, BLK, 0, stream>>>(h, L[l].gcn_b, st, n);
    k_gn_update<<<blocks(nCH), BLK, 0, stream>>>(x, h, L[l].gcn_b, st,
                                                 L[l].gn_w, L[l].gn_b, L[l].gn_ms, n);
    // GAT branch
    k_gemm128<<<mb, BLK, 0, stream>>>(x, L[l].gat_W, t, n);
    k_gat_scores<<<blocks(n), BLK, 0, stream>>>(t, L[l].att_src, L[l].att_dst, sv, dv, n);
    k_fill<<<blocks(n), BLK, 0, stream>>>(emax, n, -3.402823466e38f);
    k_edge_max<<<egB, BLK, 0, stream>>>(ei, sv, dv, emax, E, n);
    k_zero<<<blocks(n), BLK, 0, stream>>>(esum, n);
    k_edge_exp<<<egB, BLK, 0, stream>>>(ei, sv, dv, emax, esum, exb, E, n);
    k_zero<<<blocks(nCH), BLK, 0, stream>>>(h, nCH);
    k_agg_att<<<blocks(eth), BLK, 0, stream>>>(ei, t, exb, esum, h, E, n);
    k_zero<<<1, BLK, 0, stream>>>(st, 2 * CH);
    k_gn_stats<<<480, BLK, 0, stream>>>(h, L[l].gat_b, st, n);
    k_gn_update<<<blocks(nCH), BLK, 0, stream>>>(x, h, L[l].gat_b, st,
                                                 L[l].gn_w, L[l].gn_b, L[l].gn_ms, n);
  }

  // --- output head: xg = lrelu(gcn(x @ cov_out_W) + b) ---
  k_dot_out<<<blocks(n), BLK, 0, stream>>>(x, cov_out_W, t, n);
  k_zero<<<blocks(n), BLK, 0, stream>>>(xg, n);
  k_agg_scalar<<<egB, BLK, 0, stream>>>(ei, ew, dinv, t, xg, E, n);
  k_finalize_xg<<<blocks(n), BLK, 0, stream>>>(xg, cov_out_b, n);

  // --- FC head ---
  k_zero<<<1, BLK, 0, stream>>>(hfc, CH);
  k_fc1<<<(unsigned)((n + 511) / 512), BLK, 0, stream>>>(xg, fc_W1, hfc, n);
  k_fc2<<<blocks(plen), BLK, 0, stream>>>(hfc, fc_b1, fc_W2, fc_b2, (float*)d_out, plen);
}